// CannyEdgeDetector_13254269075813
// MI455X (gfx1250) — compile-verified
//
#include <hip/hip_runtime.h>
#include <math.h>

#define HH 1024
#define WW 1024
#define NIMG 8
#define HWSZ (HH * WW)

typedef __attribute__((ext_vector_type(2))) float v2f;
typedef __attribute__((ext_vector_type(8))) float v8f;

__device__ __forceinline__ int reflect_idx(int i, int n) {
    if (i < 0) i = -i;
    if (i >= n) i = 2 * n - 2 - i;
    return i;
}

__device__ __forceinline__ float sigmoid100(float x) {
    return 1.0f / (1.0f + expf(-100.0f * x));
}

// 5-tap normalized Gaussian (sigma=1): g[d] for d in [0,4], else 0.
// Branch-free select chain -> v_cmp/v_cndmask, no memory, no EXEC games.
__device__ __forceinline__ float gband(int d) {
    unsigned u = (unsigned)d;
    float r = (u == 2u) ? 0.402619946f
            : ((u == 1u) | (u == 3u)) ? 0.244201342f
            : ((u == 0u) | (u == 4u)) ? 0.054488685f
            : 0.0f;
    return r;
}

// ---------------------------------------------------------------------------
// Kernel 1: separable 5x5 Gaussian blur as banded matmuls on the WMMA pipe.
// One wave32 computes one 16x16 output tile.
//   H(20x16)  = X(20x20)  * Wh(20x16),  Wh[k][n] = g[k-n]
//   Out(16x16)= Wv(16x20) * H(20x16),   Wv[m][k] = g[k-m]
// K=20 -> 5 chained V_WMMA_F32_16X16X4_F32 per matmul (15 total).
// The per-lane weight fragment for chunk kk is IDENTICAL for the horizontal
// B-matrix and the vertical A-matrix, so it is computed once into registers.
// ---------------------------------------------------------------------------
__global__ __launch_bounds__(256) void canny_blur_wmma(const float* __restrict__ src,
                                                       float* __restrict__ dst) {
    constexpr int XS = 21;  // X tile row stride (20 cols + 1 pad, bank stagger)
    constexpr int HS = 17;  // H tile row stride (16 cols + 1 pad)
    __shared__ float lds[8 * (20 * XS + 20 * HS)];

    const int wave = threadIdx.x >> 5;
    const int lane = threadIdx.x & 31;
    float* Xs = lds + wave * (20 * XS + 20 * HS);
    float* Hs = Xs + 20 * XS;

    const int tileId = blockIdx.x * 8 + wave;   // 32768 tiles total
    const int img = tileId >> 12;               // 4096 tiles per image
    const int rem = tileId & 4095;
    const int tyo = (rem >> 6) * 16;
    const int txo = (rem & 63) * 16;

    const float* sp = src + (size_t)img * HWSZ;

    // Stage 20x20 input tile (halo 2, reflect padding).
    for (int idx = lane; idx < 400; idx += 32) {
        int r = idx / 20, c = idx - r * 20;
        int gr = reflect_idx(tyo + r - 2, HH);
        int gc = reflect_idx(txo + c - 2, WW);
        Xs[r * XS + c] = sp[gr * WW + gc];
    }
    __syncthreads();

    const int m    = lane & 15;   // M (or N) index within 16
    const int hi   = lane >> 4;   // half-wave select
    const int koff = hi * 2;      // fp32 A/B frag: lanes 16-31 hold K+2/K+3

    // Per-lane banded weight fragments, one per K-chunk; register resident.
    v2f wf[5];
#pragma unroll
    for (int kk = 0; kk < 5; ++kk) {
        wf[kk].x = gband(kk * 4 + koff     - m);
        wf[kk].y = gband(kk * 4 + koff + 1 - m);
    }

    // ---- Horizontal pass: two 16-row groups cover the 20 input rows.
#pragma unroll
    for (int grp = 0; grp < 2; ++grp) {
        v8f acc = {};
        int row = grp * 16 + m;
        if (row > 19) row = 19;                     // dead rows of group 1
        const float* xr = Xs + row * XS + koff;
#pragma unroll
        for (int kk = 0; kk < 5; ++kk) {
            v2f a;
            a.x = xr[kk * 4];
            a.y = xr[kk * 4 + 1];
            acc = __builtin_amdgcn_wmma_f32_16x16x4_f32(
                false, a, false, wf[kk], (short)0, acc, false, false);
        }
        if (grp == 0) {                             // rows 0..15, all valid
#pragma unroll
            for (int r = 0; r < 8; ++r)             // C/D: M = r + 8*hi
                Hs[(r + 8 * hi) * HS + m] = acc[r];
        } else {                                    // rows 16..19: hi==0, r<4
            if (lane < 16) {
#pragma unroll
                for (int r = 0; r < 4; ++r)
                    Hs[(16 + r) * HS + m] = acc[r];
            }
        }
    }
    __syncthreads();

    // ---- Vertical pass: A = weights (same fragments), B = H from LDS.
    v8f acc = {};
#pragma unroll
    for (int kk = 0; kk < 5; ++kk) {
        v2f b;
        b.x = Hs[(kk * 4 + koff)     * HS + m];
        b.y = Hs[(kk * 4 + koff + 1) * HS + m];
        acc = __builtin_amdgcn_wmma_f32_16x16x4_f32(
            false, wf[kk], false, b, (short)0, acc, false, false);
    }
    float* dp = dst + (size_t)img * HWSZ;
#pragma unroll
    for (int r = 0; r < 8; ++r) {
        int orow = tyo + r + 8 * hi;
        dp[orow * WW + txo + m] = acc[r];
    }
}

// ---------------------------------------------------------------------------
// Kernel 2: Sobel + magnitude + direction + non-max suppression + min/max.
// Block = 256 threads, 32x8 output tile.
// ---------------------------------------------------------------------------
__global__ __launch_bounds__(256) void canny_sobel_nms(const float* __restrict__ sm,
                                                       float* __restrict__ sup,
                                                       unsigned* __restrict__ mm) {
    constexpr int SS = 37;  // smoothed tile: 12 rows x 36 cols (halo 2)
    constexpr int GS = 35;  // grad tile:     10 rows x 34 cols (halo 1)
    __shared__ float S[12 * SS];
    __shared__ float Gt[10 * GS];
    __shared__ float red[16];

    const int tid = threadIdx.x;
    const int img = blockIdx.z;
    const int rb = blockIdx.y * 8;
    const int cb = blockIdx.x * 32;
    const float* sp = sm + (size_t)img * HWSZ;

    for (int idx = tid; idx < 12 * 36; idx += 256) {
        int r = idx / 36, c = idx - r * 36;
        S[r * SS + c] = sp[reflect_idx(rb + r - 2, HH) * WW +
                           reflect_idx(cb + c - 2, WW)];
    }
    __syncthreads();

    for (int idx = tid; idx < 10 * 34; idx += 256) {
        int r = idx / 34, c = idx - r * 34;
        int rs = r + 1, cs = c + 1;
        float gx = -S[(rs - 1) * SS + cs - 1] + S[(rs - 1) * SS + cs + 1]
                 - 2.0f * S[rs * SS + cs - 1] + 2.0f * S[rs * SS + cs + 1]
                 - S[(rs + 1) * SS + cs - 1] + S[(rs + 1) * SS + cs + 1];
        float gy =  S[(rs - 1) * SS + cs - 1] + 2.0f * S[(rs - 1) * SS + cs]
                 +  S[(rs - 1) * SS + cs + 1]
                 -  S[(rs + 1) * SS + cs - 1] - 2.0f * S[(rs + 1) * SS + cs]
                 -  S[(rs + 1) * SS + cs + 1];
        Gt[r * GS + c] = sqrtf(gx * gx + gy * gy);
    }
    __syncthreads();

    const int tx = tid & 31;
    const int ty = tid >> 5;
    const int rs = ty + 2, cs = tx + 2;  // center in S coords
    const int rg = ty + 1, cg = tx + 1;  // center in G coords

    float gx = -S[(rs - 1) * SS + cs - 1] + S[(rs - 1) * SS + cs + 1]
             - 2.0f * S[rs * SS + cs - 1] + 2.0f * S[rs * SS + cs + 1]
             - S[(rs + 1) * SS + cs - 1] + S[(rs + 1) * SS + cs + 1];
    float gy =  S[(rs - 1) * SS + cs - 1] + 2.0f * S[(rs - 1) * SS + cs]
             +  S[(rs - 1) * SS + cs + 1]
             -  S[(rs + 1) * SS + cs - 1] - 2.0f * S[(rs + 1) * SS + cs]
             -  S[(rs + 1) * SS + cs + 1];

    float dir = atan2f(gy, gx) * 57.29577951308232f;
    if (dir < 0.0f) dir += 180.0f;

    float g  = Gt[rg * GS + cg];
    float e_ = Gt[rg * GS + cg + 1],       w_ = Gt[rg * GS + cg - 1];
    float n_ = Gt[(rg - 1) * GS + cg],     s_ = Gt[(rg + 1) * GS + cg];
    float ne = Gt[(rg - 1) * GS + cg + 1], sw = Gt[(rg + 1) * GS + cg - 1];
    float nw = Gt[(rg - 1) * GS + cg - 1], se = Gt[(rg + 1) * GS + cg + 1];

    float v = 0.0f;
    if (((dir < 22.5f) || (dir >= 157.5f)) && (g > e_) && (g > w_)) v = g;
    if ((dir >= 22.5f)  && (dir < 67.5f)  && (g > sw) && (g > ne)) v = g;
    if ((dir >= 67.5f)  && (dir < 112.5f) && (g > n_) && (g > s_)) v = g;
    if ((dir >= 112.5f) && (dir < 157.5f) && (g > se) && (g > nw)) v = g;

    sup[(size_t)img * HWSZ + (rb + ty) * WW + cb + tx] = v;

    // Block min/max (sup >= 0, so float bits compare monotonically as uint).
    float mnv = v, mxv = v;
#pragma unroll
    for (int o = 16; o > 0; o >>= 1) {
        mnv = fminf(mnv, __shfl_xor(mnv, o));
        mxv = fmaxf(mxv, __shfl_xor(mxv, o));
    }
    if (tx == 0) { red[ty] = mnv; red[8 + ty] = mxv; }
    __syncthreads();
    if (tid == 0) {
        float a = red[0], b = red[8];
#pragma unroll
        for (int i = 1; i < 8; ++i) {
            a = fminf(a, red[i]);
            b = fmaxf(b, red[8 + i]);
        }
        atomicMin(&mm[0], __float_as_uint(a));
        atomicMax(&mm[1], __float_as_uint(b));
    }
}

// ---------------------------------------------------------------------------
// Kernel 3: soft thresholds + 3x3 hysteresis conv + final mask.
// ---------------------------------------------------------------------------
__global__ __launch_bounds__(256) void canny_hyst(const float* __restrict__ sup,
                                                  const unsigned* __restrict__ mm,
                                                  float* __restrict__ out) {
    constexpr int TS = 35;  // strong tile: 10 rows x 34 cols (halo 1)
    __shared__ float ST[10 * TS];

    const int tid = threadIdx.x;
    const int img = blockIdx.z;
    const int rb = blockIdx.y * 8;
    const int cb = blockIdx.x * 32;

    const float mn = __uint_as_float(mm[0]);
    const float mx = __uint_as_float(mm[1]);
    const float low  = mn + (mx - mn) * 0.1f;
    const float high = mn + (mx - mn) * 0.25f;
    const float* sp = sup + (size_t)img * HWSZ;

    for (int idx = tid; idx < 10 * 34; idx += 256) {
        int r = idx / 34, c = idx - r * 34;
        float sv = sp[reflect_idx(rb + r - 1, HH) * WW +
                      reflect_idx(cb + c - 1, WW)];
        ST[r * TS + c] = sigmoid100(sv - high);
    }
    __syncthreads();

    const int tx = tid & 31;
    const int ty = tid >> 5;
    const int r = ty + 1, c = tx + 1;

    float strongC = ST[r * TS + c];
    float hs = 0.0f;
#pragma unroll
    for (int dr = -1; dr <= 1; ++dr)
#pragma unroll
        for (int dc = -1; dc <= 1; ++dc)
            hs += ST[(r + dr) * TS + c + dc];

    float supC = sp[(rb + ty) * WW + cb + tx];
    float weak = sigmoid100(supC - low) * (1.0f - strongC);
    float mask = strongC + weak * sigmoid100(hs - 0.5f);
    // Final output: written once, never re-read on device -> non-temporal
    // store keeps the 32 MiB result from displacing L2-resident tiles.
    __builtin_nontemporal_store(mask,
        &out[(size_t)img * HWSZ + (rb + ty) * WW + cb + tx]);
}

__global__ void canny_init(unsigned* __restrict__ mm) {
    mm[0] = 0x7F800000u;  // +inf  (min accumulator)
    mm[1] = 0u;           // +0.0f (max accumulator; sup >= 0)
}

extern "C" void kernel_launch(void* const* d_in, const int* in_sizes, int n_in,
                              void* d_out, int out_size, void* d_ws, size_t ws_size,
                              hipStream_t stream) {
    (void)in_sizes; (void)n_in; (void)out_size; (void)ws_size;

    const float* img = (const float*)d_in[0];
    float* smoothed = (float*)d_ws;
    float* sup = smoothed + (size_t)NIMG * HWSZ;
    unsigned* mm = (unsigned*)(sup + (size_t)NIMG * HWSZ);
    float* out = (float*)d_out;

    canny_init<<<1, 1, 0, stream>>>(mm);

    // 8 images * 64*64 tiles = 32768 tiles; 8 waves/block -> 4096 blocks.
    canny_blur_wmma<<<4096, 256, 0, stream>>>(img, smoothed);

    dim3 grid(WW / 32, HH / 8, NIMG);
    canny_sobel_nms<<<grid, 256, 0, stream>>>(smoothed, sup, mm);
    canny_hyst<<<grid, 256, 0, stream>>>(sup, mm, out);
}